// LambdaNDCGLoss_54348516163757
// MI455X (gfx1250) — compile-verified
//
#include <hip/hip_runtime.h>
#include <hip/hip_bf16.h>

typedef __attribute__((ext_vector_type(2))) float v2f;
typedef __attribute__((ext_vector_type(8))) float v8f;

#define WMMA4(A, B, C) \
    __builtin_amdgcn_wmma_f32_16x16x4_f32(false, (A), false, (B), (short)0, (C), false, false)

#define B_ROWS 32
#define N_ITEMS 1024
#define BANDS 8           // i-tile bands per row
#define SUB_I 8           // i-tiles per band (8*8*16 = 1024)
#define JTILES 64         // N/16
#define BLOCK 256

#define LOG2E 1.44269504088896340736f
#define LN2   0.69314718055994530942f

// One workgroup per (band, batch-row). Stages the row into LDS, scans the mask
// to build the NDCG discount, then builds 16x16 pair tiles with f32 WMMAs.
// Per i-tile (hoisted out of the j-loop):
//   Ti = p_i * ones, Tt = t_i * ones        (2 WMMAs per 8 j-tiles)
// Per (i,j) tile (inner loop, 5 WMMAs):
//   pred_diff = Ti + 1*(-p_j)
//   true_diff = Tt + 1*(-t_j)
//   W         = disc_i*m_j + m_i*disc_j  == (disc_i+disc_j)*m_i*m_j
//   P         = m_i*m_j
// Elementwise: loss += softplus(-pred_diff)*(true_diff>0)*W ; cnt += (true_diff>0)*P
// All WMMAs are K-permutation-proof: one operand is constant across K (ones or
// replicated), the other has exactly one nonzero K slot.
__global__ __launch_bounds__(BLOCK) void lambda_ndcg_pairs(
    const float* __restrict__ y_pred, const float* __restrict__ y_true,
    const int* __restrict__ masks, float* __restrict__ ws) {
    __shared__ float s_pred[N_ITEMS];
    __shared__ float s_true[N_ITEMS];
    __shared__ float s_m[N_ITEMS];
    __shared__ float s_disc[N_ITEMS];
    __shared__ float s_red[BLOCK];
    __shared__ float s_red2[BLOCK];

    const int band = blockIdx.x;       // 0..7
    const int b    = blockIdx.y;       // 0..31
    const int tid  = threadIdx.x;

    // ---- stage row into LDS (128-bit loads: 4 contiguous elems per thread) ----
    const float4* gp4 = (const float4*)(y_pred + (size_t)b * N_ITEMS);
    const float4* gt4 = (const float4*)(y_true + (size_t)b * N_ITEMS);
    const int4*   gm4 = (const int4*)(masks  + (size_t)b * N_ITEMS);
    {
        const float4 p4 = gp4[tid];
        const float4 t4 = gt4[tid];
        const int4   m4 = gm4[tid];
        const int i4 = tid * 4;
        ((float4*)s_pred)[tid] = p4;
        ((float4*)s_true)[tid] = t4;
        s_m[i4 + 0] = (m4.x > 0) ? 1.0f : 0.0f;
        s_m[i4 + 1] = (m4.y > 0) ? 1.0f : 0.0f;
        s_m[i4 + 2] = (m4.z > 0) ? 1.0f : 0.0f;
        s_m[i4 + 3] = (m4.w > 0) ? 1.0f : 0.0f;
    }
    __syncthreads();

    // ---- cumsum(mask) scan -> discount  disc[i] = m ? 1/log2(cs+1) : 0 ----
    const int base = tid * 4;
    const float m0 = s_m[base + 0], m1 = s_m[base + 1];
    const float m2 = s_m[base + 2], m3 = s_m[base + 3];
    const float c1 = m0, c2 = m0 + m1, c3 = c2 + m2, c4 = c3 + m3;
    s_red[tid] = c4;
    __syncthreads();
    for (int off = 1; off < BLOCK; off <<= 1) {
        float v = (tid >= off) ? s_red[tid - off] : 0.0f;
        __syncthreads();
        s_red[tid] += v;
        __syncthreads();
    }
    const float excl = s_red[tid] - c4;
    s_disc[base + 0] = (m0 > 0.0f) ? 1.0f / log2f(excl + c1 + 1.0f) : 0.0f;
    s_disc[base + 1] = (m1 > 0.0f) ? 1.0f / log2f(excl + c2 + 1.0f) : 0.0f;
    s_disc[base + 2] = (m2 > 0.0f) ? 1.0f / log2f(excl + c3 + 1.0f) : 0.0f;
    s_disc[base + 3] = (m3 > 0.0f) ? 1.0f / log2f(excl + c4 + 1.0f) : 0.0f;
    __syncthreads();

    // ---- per-wave 16x16 tile loop (wave32) ----
    const int lane     = tid & 31;
    const int wave     = tid >> 5;          // 0..7
    const int col      = lane & 15;
    const bool lowHalf = (lane < 16);

    float accL = 0.0f, accC = 0.0f;

    v2f ones2;  ones2.x = 1.0f; ones2.y = 1.0f;
    const v8f zero8 = {};

    for (int itl = 0; itl < SUB_I; ++itl) {
        const int i0 = (band * SUB_I + itl) * 16;
        const float pi = s_pred[i0 + col], ti = s_true[i0 + col];
        const float mi = s_m[i0 + col],    di = s_disc[i0 + col];

        // "single" operands: exactly one nonzero K slot; "replicated": all K slots.
        v2f aS_p;  aS_p.x  = lowHalf ? pi  : 0.0f; aS_p.y  = 0.0f;
        v2f aS_t;  aS_t.x  = lowHalf ? ti  : 0.0f; aS_t.y  = 0.0f;
        v2f aS_d;  aS_d.x  = lowHalf ? di  : 0.0f; aS_d.y  = 0.0f;
        v2f aS_m;  aS_m.x  = lowHalf ? mi  : 0.0f; aS_m.y  = 0.0f;
        v2f aR_m;  aR_m.x  = mi;  aR_m.y = mi;

        // Hoisted stage-1 broadcasts (depend only on i); kept live as WMMA C
        // inputs across the whole j-loop (VDST != SRC2, so no copies needed).
        const v8f Ti = WMMA4(aS_p, ones2, zero8);   // p_i broadcast over j
        const v8f Tt = WMMA4(aS_t, ones2, zero8);   // t_i broadcast over j

        for (int jt = wave; jt < JTILES; jt += 8) {
            const int j0 = jt * 16;
            const float pj = s_pred[j0 + col], tj = s_true[j0 + col];
            const float mj = s_m[j0 + col],    dj = s_disc[j0 + col];

            v2f bS_np; bS_np.x = lowHalf ? -pj : 0.0f; bS_np.y = 0.0f;
            v2f bS_nt; bS_nt.x = lowHalf ? -tj : 0.0f; bS_nt.y = 0.0f;
            v2f bS_d;  bS_d.x  = lowHalf ? dj  : 0.0f; bS_d.y  = 0.0f;
            v2f bR_m;  bR_m.x  = mj;  bR_m.y = mj;

            const v8f Dp = WMMA4(ones2, bS_np, Ti);   // p_i - p_j
            const v8f Dt = WMMA4(ones2, bS_nt, Tt);   // t_i - t_j
            v8f Wm       = WMMA4(aS_d, bR_m, zero8);  // disc_i * m_j
            Wm           = WMMA4(aR_m, bS_d, Wm);     // + m_i * disc_j
            const v8f Pm = WMMA4(aS_m, bR_m, zero8);  // m_i * m_j

#pragma unroll
            for (int e = 0; e < 8; ++e) {
                // softplus(z) = ln2*(max(z2,0) + log2(1 + 2^-|z2|)), z2 = z*log2e
                const float z2 = -Dp[e] * LOG2E;              // sigma = 1
                const float t  = __builtin_amdgcn_exp2f(-fabsf(z2));   // v_exp_f32
                const float l2 = __builtin_amdgcn_logf(1.0f + t);      // v_log_f32
                const float sp = LN2 * (fmaxf(z2, 0.0f) + l2);
                const float g  = (Dt[e] > 0.0f) ? 1.0f : 0.0f;
                accL += g * Wm[e] * sp;
                accC += g * Pm[e];
            }
        }
    }

    // ---- block reduction, write unique partial slot (deterministic) ----
    s_red[tid]  = accL;
    s_red2[tid] = accC;
    __syncthreads();
    for (int s = BLOCK / 2; s > 0; s >>= 1) {
        if (tid < s) {
            s_red[tid]  += s_red[tid + s];
            s_red2[tid] += s_red2[tid + s];
        }
        __syncthreads();
    }
    if (tid == 0) {
        ws[b * BANDS + band]                  = s_red[0];   // loss partial
        ws[B_ROWS * BANDS + b * BANDS + band] = s_red2[0];  // count partial
    }
}

__global__ void lambda_ndcg_finalize(const float* __restrict__ ws,
                                     float* __restrict__ out) {
    __shared__ float sr[B_ROWS];
    const int t = threadIdx.x;  // 32 threads, one per batch row
    float L = 0.0f, C = 0.0f;
    for (int k = 0; k < BANDS; ++k) {
        L += ws[t * BANDS + k];
        C += ws[B_ROWS * BANDS + t * BANDS + k];
    }
    sr[t] = L / (C + 1e-12f);
    __syncthreads();
    for (int s = B_ROWS / 2; s > 0; s >>= 1) {
        if (t < s) sr[t] += sr[t + s];
        __syncthreads();
    }
    if (t == 0) out[0] = sr[0] / (float)B_ROWS;
}

extern "C" void kernel_launch(void* const* d_in, const int* in_sizes, int n_in,
                              void* d_out, int out_size, void* d_ws, size_t ws_size,
                              hipStream_t stream) {
    (void)in_sizes; (void)n_in; (void)out_size; (void)ws_size;
    const float* y_pred = (const float*)d_in[0];
    const float* y_true = (const float*)d_in[1];
    const int*   masks  = (const int*)d_in[2];
    float* ws  = (float*)d_ws;   // [32*8 loss partials][32*8 count partials]
    float* out = (float*)d_out;  // scalar f32

    dim3 grid(BANDS, B_ROWS);
    lambda_ndcg_pairs<<<grid, BLOCK, 0, stream>>>(y_pred, y_true, masks, ws);
    lambda_ndcg_finalize<<<1, B_ROWS, 0, stream>>>(ws, out);
}